// CausalGRNEMA_30477087932471
// MI455X (gfx1250) — compile-verified
//
#include <hip/hip_runtime.h>
#include <math.h>

// ---------------------------------------------------------------------------
// CausalGRNEMA on gfx1250 (MI455X): EMA scan as 16x16 lower-triangular GEMM
// via V_WMMA_F32_16X16X4_F32 + async direct-to-LDS tile staging (ASYNCcnt),
// wave32 layouts per CDNA5 ISA 7.12.2.
// ---------------------------------------------------------------------------

typedef __attribute__((ext_vector_type(2))) float v2f;
typedef __attribute__((ext_vector_type(8))) float v8f;

#define ALPHA    0.99f
#define EPS      1e-6f
#define EMA_INIT 1e-4f
#define L2A      (-0.0144995696951151f)   // log2(0.99)

#define Bn    16
#define Tn    8192
#define Cn    512
#define NCH   64              // chunks along T
#define LCH   (Tn / NCH)      // 128 timesteps per chunk
#define ITERS (LCH / 16)      // 8 WMMA tiles per chunk

// ---- wave32 half-wave (16-lane) reduction via ds_swizzle (group-of-32) ----
template <int IMM>
__device__ __forceinline__ float swz(float v) {
  return __int_as_float(__builtin_amdgcn_ds_swizzle(__float_as_int(v), IMM));
}
__device__ __forceinline__ float halfsum16(float v) {
  v += swz<0x041F>(v);   // xor 1
  v += swz<0x081F>(v);   // xor 2
  v += swz<0x101F>(v);   // xor 4
  v += swz<0x201F>(v);   // xor 8  (stays within 16-lane half)
  return v;
}

// ---- gfx1250 async global->LDS (ASYNCcnt) ---------------------------------
__device__ __forceinline__ void async_ld_b128(unsigned lds_off, unsigned long long gaddr) {
  asm volatile("global_load_async_to_lds_b128 %0, %1, off"
               :: "v"(lds_off), "v"(gaddr) : "memory");
}
__device__ __forceinline__ void wait_async0() {
  asm volatile("s_wait_asynccnt 0" ::: "memory");
}

// ---- kernel 0: bias-correction table rdenom[g] = 1/(1 - alpha^(g+1) + eps) -
__global__ void k_rdenom(float* __restrict__ rdenom) {
  int g = blockIdx.x * blockDim.x + threadIdx.x;
  if (g < Tn) {
    float at = exp2f((float)(g + 1) * L2A);
    rdenom[g] = 1.0f / ((1.0f - at) + EPS);
  }
}

// ---- kernel 1: per-chunk alpha-weighted partials (Horner), b128 coalesced --
__global__ __launch_bounds__(128) void k_partial(const float* __restrict__ x,
                                                 float* __restrict__ Sp) {
  const int b  = blockIdx.x / NCH;
  const int ch = blockIdx.x % NCH;
  const int c4 = threadIdx.x * 4;
  const float* xp = x + ((size_t)b * Tn + (size_t)ch * LCH) * Cn + c4;
  float4 S = make_float4(0.f, 0.f, 0.f, 0.f);
#pragma unroll 8
  for (int j = 0; j < LCH; ++j) {
    float4 v = *(const float4*)(xp + (size_t)j * Cn);
    S.x = fmaf(ALPHA, S.x, (1.0f - ALPHA) * v.x * v.x);
    S.y = fmaf(ALPHA, S.y, (1.0f - ALPHA) * v.y * v.y);
    S.z = fmaf(ALPHA, S.z, (1.0f - ALPHA) * v.z * v.z);
    S.w = fmaf(ALPHA, S.w, (1.0f - ALPHA) * v.w * v.w);
  }
  *(float4*)&Sp[((size_t)b * NCH + ch) * Cn + c4] = S;
}

// ---- kernel 2: exact carry scan across chunk boundaries --------------------
__global__ __launch_bounds__(512) void k_carry(const float* __restrict__ Sp,
                                               float* __restrict__ carry) {
  const int idx = blockIdx.x * blockDim.x + threadIdx.x;   // 0..B*C-1
  const int b = idx >> 9, c = idx & 511;
  const float powAL = exp2f((float)LCH * L2A);             // alpha^LCH
  float s = EMA_INIT;                                      // state before t=1
  for (int ch = 0; ch < NCH; ++ch) {
    size_t o = ((size_t)b * NCH + ch) * Cn + c;
    carry[o] = s;
    s = fmaf(powAL, s, Sp[o]);
  }
}

// ---- kernel 3: main fused pass --------------------------------------------
// Block = (b, chunk): 512 threads = 16 waves, each wave owns 32 channels
// (two 16-wide WMMA column tiles). 16-timestep tile per iteration, with the
// next tile's 32KB x-slab async-loaded to LDS while the current one computes.
__global__ __launch_bounds__(512) void k_main(const float* __restrict__ x,
                                              const float* __restrict__ gamma,
                                              const float* __restrict__ beta,
                                              const float* __restrict__ rdenom,
                                              const float* __restrict__ carry,
                                              float* __restrict__ y) {
  __shared__ float xs[2][16 * Cn];    // 2 x 32 KB double-buffered x tiles
  __shared__ float part[16][16];      // per-wave row partial sums
  __shared__ float minv[16];          // 1/(mean_g + eps) per row

  const int b    = blockIdx.x / NCH;
  const int ch   = blockIdx.x % NCH;
  const int tid  = threadIdx.x;
  const int wave = tid >> 5;
  const int lane = tid & 31;
  const int hi   = lane >> 4;         // half-wave select
  const int col  = lane & 15;         // WMMA column (channel) / A-row
  const int c0   = wave * 32 + col;   // channel tile 0
  const int c1   = c0 + 16;           // channel tile 1

  // A = lower-triangular powers of alpha, M[r][k] = alpha^(r-k), r = col.
  // 32-bit A 16x4 layout: lanes 0-15 -> {K0,K1}, lanes 16-31 -> {K2,K3}.
  v2f Am[4];
#pragma unroll
  for (int kb = 0; kb < 4; ++kb)
#pragma unroll
    for (int vi = 0; vi < 2; ++vi) {
      int k = 4 * kb + 2 * hi + vi;
      int d = col - k;
      Am[kb][vi] = (d >= 0) ? exp2f((float)d * L2A) : 0.0f;
    }
  // alpha^(row+1) for folding the incoming carry; row = v + 8*hi.
  float apows[8];
#pragma unroll
  for (int v = 0; v < 8; ++v) apows[v] = exp2f((float)(v + 8 * hi + 1) * L2A);

  const float gm0 = gamma[c0], gm1 = gamma[c1];
  const float bt0 = beta[c0],  bt1 = beta[c1];

  const size_t cb = ((size_t)b * NCH + ch) * Cn;
  float s0 = carry[cb + c0];
  float s1 = carry[cb + c1];

  const int t0base = ch * LCH;
  const float* xbase = x + ((size_t)b * Tn + t0base) * Cn;

  // async LDS staging addresses: 2048 x 16B per tile, lanes contiguous
  const unsigned lds0 = ((unsigned)(size_t)(&xs[0][0])) + (unsigned)tid * 16u;
  const unsigned lds1 = ((unsigned)(size_t)(&xs[1][0])) + (unsigned)tid * 16u;

  // prologue: stage tile 0 into buffer 0
  {
    unsigned long long g = (unsigned long long)(size_t)xbase + (size_t)tid * 16u;
#pragma unroll
    for (int q = 0; q < 4; ++q) async_ld_b128(lds0 + q * 8192u, g + q * 8192u);
  }
  wait_async0();
  __syncthreads();

  for (int it = 0; it < ITERS; ++it) {
    const int cur = it & 1;
    const int t0  = t0base + it * 16;
    const float* xt = xbase + (size_t)it * 16 * Cn;
    float*       yt = y + ((size_t)b * Tn + t0) * Cn;
    const float* xsc = xs[cur];

    // kick off next tile's async staging into the other buffer
    if (it + 1 < ITERS) {
      unsigned long long g = (unsigned long long)(size_t)(xt + 16 * Cn) + (size_t)tid * 16u;
      unsigned ldsn = cur ? lds0 : lds1;
#pragma unroll
      for (int q = 0; q < 4; ++q) async_ld_b128(ldsn + q * 8192u, g + q * 8192u);
    }

    // B operand: b[k][n] = (1-a)*x^2. 4x16 f32 B layout mirrors A:
    // lanes 0-15 -> {K0,K1}, lanes 16-31 -> {K2,K3}, N = col.
    v2f Bm0[4], Bm1[4];
#pragma unroll
    for (int kb = 0; kb < 4; ++kb)
#pragma unroll
      for (int vi = 0; vi < 2; ++vi) {
        int k = 4 * kb + 2 * hi + vi;
        float xv0 = xsc[k * Cn + c0];
        float xv1 = xsc[k * Cn + c1];
        Bm0[kb][vi] = (1.0f - ALPHA) * xv0 * xv0;
        Bm1[kb][vi] = (1.0f - ALPHA) * xv1 * xv1;
      }

    // 16x16 tile of the scan: D = M @ B  (4 chained K=4 f32 WMMAs per tile)
    v8f D0 = {};
    v8f D1 = {};
#pragma unroll
    for (int kb = 0; kb < 4; ++kb) {
      D0 = __builtin_amdgcn_wmma_f32_16x16x4_f32(false, Am[kb], false, Bm0[kb],
                                                 (short)0, D0, false, false);
      D1 = __builtin_amdgcn_wmma_f32_16x16x4_f32(false, Am[kb], false, Bm1[kb],
                                                 (short)0, D1, false, false);
    }

    // ema[row] = D[row] + alpha^(row+1) * carry
    float e0[8], e1[8];
#pragma unroll
    for (int v = 0; v < 8; ++v) {
      e0[v] = D0[v] + apows[v] * s0;
      e1[v] = D1[v] + apows[v] * s1;
    }

    // Carry to next tile = ema at row 15 (hi half, vgpr 7); broadcast with
    // ds_bpermute so lane n and n+16 both get channel n's carry.
    s0 = __int_as_float(__builtin_amdgcn_ds_bpermute((col + 16) << 2,
                                                     __float_as_int(e0[7])));
    s1 = __int_as_float(__builtin_amdgcn_ds_bpermute((col + 16) << 2,
                                                     __float_as_int(e1[7])));

    // bias correction: 16 uniform values -> scalar loads + per-lane cndmask
    const float* rdt = rdenom + t0;    // block-uniform pointer
    float g0[8], g1[8], rs[8];
#pragma unroll
    for (int v = 0; v < 8; ++v) {
      float rl = rdt[v];
      float rh = rdt[v + 8];
      float r  = hi ? rh : rl;
      // raw v_sqrt_f32: argument >= EPS, no denormal fixup needed
      g0[v] = __builtin_amdgcn_sqrtf(fmaf(e0[v], r, EPS));
      g1[v] = __builtin_amdgcn_sqrtf(fmaf(e1[v], r, EPS));
      rs[v] = halfsum16(g0[v] + g1[v]);
    }

    if (col == 0) {
#pragma unroll
      for (int v = 0; v < 8; ++v) part[wave][8 * hi + v] = rs[v];
    }
    __syncthreads();
    if (tid < 16) {
      float tot = 0.0f;
#pragma unroll
      for (int w = 0; w < 16; ++w) tot += part[w][tid];
      minv[tid] = __builtin_amdgcn_rcpf(tot * (1.0f / (float)Cn) + EPS);
    }
    __syncthreads();

    float mi[8];
#pragma unroll
    for (int v = 0; v < 8; ++v) mi[v] = minv[8 * hi + v];

    // y = gamma*(x*n) + beta + x  (x reused from LDS, no second HBM read)
#pragma unroll
    for (int v = 0; v < 8; ++v) {
      int rg = v + 8 * hi;
      float xv0 = xsc[rg * Cn + c0];
      float xv1 = xsc[rg * Cn + c1];
      yt[(size_t)rg * Cn + c0] = fmaf(gm0, xv0 * (g0[v] * mi[v]), bt0 + xv0);
      yt[(size_t)rg * Cn + c1] = fmaf(gm1, xv1 * (g1[v] * mi[v]), bt1 + xv1);
    }

    // next buffer fully resident + all waves done with current buffer
    wait_async0();
    __syncthreads();
  }
}

// ---------------------------------------------------------------------------
extern "C" void kernel_launch(void* const* d_in, const int* in_sizes, int n_in,
                              void* d_out, int out_size, void* d_ws, size_t ws_size,
                              hipStream_t stream) {
  const float* x     = (const float*)d_in[0];
  const float* gamma = (const float*)d_in[1];
  const float* beta  = (const float*)d_in[2];
  float*       y     = (float*)d_out;

  // workspace layout (floats): [rdenom Tn][Sp B*NCH*C][carry B*NCH*C] ~4.2 MB
  float* w      = (float*)d_ws;
  float* rdenom = w;
  float* Sp     = w + Tn;
  float* carry  = Sp + (size_t)Bn * NCH * Cn;

  k_rdenom <<<Tn / 256, 256, 0, stream>>>(rdenom);
  k_partial<<<Bn * NCH, 128, 0, stream>>>(x, Sp);
  k_carry  <<<(Bn * Cn) / 512, 512, 0, stream>>>(Sp, carry);
  k_main   <<<Bn * NCH, 512, 0, stream>>>(x, gamma, beta, rdenom, carry, y);
}